// AttentionFusion_21234318311826
// MI455X (gfx1250) — compile-verified
//
#include <hip/hip_runtime.h>
#include <hip/hip_bf16.h>

typedef __attribute__((ext_vector_type(2))) float v2f;
typedef __attribute__((ext_vector_type(8))) float v8f;

#define D_DIM   10240
#define T_ROWS  256
#define V_ROWS  64
#define OUT_ROWS 1600           // 256 text + 64 video + 1280 fused
#define REPEATN 20
#define SCALE_F 0.00988211768802618986f   // 1/sqrt(10240)

// ---------------------------------------------------------------------------
// Kernel 0: copy text -> out rows [0,256), video -> out rows [256,320)
// ---------------------------------------------------------------------------
__global__ void af_copy_head(const float* __restrict__ text,
                             const float* __restrict__ video,
                             float* __restrict__ out) {
  const size_t D4   = D_DIM / 4;          // 2560 float4 per row
  const size_t perB = 320 * D4;
  size_t i = (size_t)blockIdx.x * blockDim.x + threadIdx.x;
  if (i >= 4 * perB) return;
  int    b    = (int)(i / perB);
  size_t rem  = i % perB;
  int    row  = (int)(rem / D4);
  size_t col4 = rem % D4;
  const float4* src;
  if (row < T_ROWS)
    src = (const float4*)(text + ((size_t)b * T_ROWS + row) * D_DIM) + col4;
  else
    src = (const float4*)(video + ((size_t)b * V_ROWS + (row - T_ROWS)) * D_DIM) + col4;
  float4* dst = (float4*)(out + ((size_t)b * OUT_ROWS + row) * D_DIM) + col4;
  *dst = *src;
}

// ---------------------------------------------------------------------------
// Kernel 1: S[b][q][t] = SCALE * sum_d V[b][q][d] * T[b][t][d]
// grid: (16 t-tiles, 4 q-tiles, 4 batches); block: 256 threads = 8 waves.
// Each wave owns a K-chunk of 1280 and accumulates a 16x16 f32 tile via
// v_wmma_f32_16x16x4_f32; waves reduce through LDS.
// ---------------------------------------------------------------------------
__global__ void af_scores(const float* __restrict__ text,
                          const float* __restrict__ video,
                          float* __restrict__ S) {
  const int nt   = blockIdx.x;            // text tile  (0..15)
  const int mt   = blockIdx.y;            // video tile (0..3)
  const int b    = blockIdx.z;
  const int lane = threadIdx.x & 31;
  const int wave = threadIdx.x >> 5;      // 0..7
  const int q0 = mt * 16, t0 = nt * 16;

  const float* Vb = video + (size_t)b * V_ROWS * D_DIM;
  const float* Tb = text  + (size_t)b * T_ROWS * D_DIM;

  const int kChunk = D_DIM / 8;           // 1280 per wave
  const int d0   = wave * kChunk;
  const int mrow = lane & 15;             // matrix row / col this lane serves
  const int koff = (lane >> 4) * 2;       // K sub-offset per ISA A/B layout

  const float* aptr = Vb + (size_t)(q0 + mrow) * D_DIM + d0 + koff;
  const float* bptr = Tb + (size_t)(t0 + mrow) * D_DIM + d0 + koff;

  v8f c = {};
#pragma unroll 8
  for (int k = 0; k < kChunk; k += 4) {
    v2f a  = *(const v2f*)(aptr);
    v2f bb = *(const v2f*)(bptr);
    c = __builtin_amdgcn_wmma_f32_16x16x4_f32(false, a, false, bb,
                                              (short)0, c, false, false);
    aptr += 4; bptr += 4;
  }

  // cross-wave reduction of the 16x16 tile (8 partials per element)
  __shared__ float red[8 * 256];
#pragma unroll
  for (int r = 0; r < 8; ++r) red[wave * 256 + r * 32 + lane] = c[r];
  __syncthreads();

  int idx = threadIdx.x;                  // 0..255 -> one (r,lane) element
  float s = 0.0f;
#pragma unroll
  for (int w = 0; w < 8; ++w) s += red[w * 256 + idx];
  s *= SCALE_F;

  int r = idx >> 5, l = idx & 31;
  int m = r + ((l >> 4) << 3);            // C/D layout: vgpr r, lanes 16-31 -> m+8
  int n = l & 15;
  S[((size_t)b * V_ROWS + q0 + m) * T_ROWS + t0 + n] = s;
}

// ---------------------------------------------------------------------------
// Kernel 2: row-wise softmax over 256 keys. grid: 256 rows (b*64+q).
// ---------------------------------------------------------------------------
__global__ void af_softmax(float* __restrict__ S) {
  const int row = blockIdx.x;
  float* p = S + (size_t)row * T_ROWS;
  const int t = threadIdx.x;
  float v = p[t];

  __shared__ float red[256];
  red[t] = v;
  __syncthreads();
#pragma unroll
  for (int s = 128; s > 0; s >>= 1) {
    if (t < s) red[t] = fmaxf(red[t], red[t + s]);
    __syncthreads();
  }
  float mx = red[0];
  __syncthreads();

  float e = __expf(v - mx);
  red[t] = e;
  __syncthreads();
#pragma unroll
  for (int s = 128; s > 0; s >>= 1) {
    if (t < s) red[t] += red[t + s];
    __syncthreads();
  }
  p[t] = e * (1.0f / red[0]);
}

// ---------------------------------------------------------------------------
// Kernel 3: attn[q][d] = sum_t W[q][t]*T[t][d]; fused = attn + video;
// write fused to the 20 replicated output rows.
// grid: (80 d-groups, 4 q-tiles, 4 batches); block 256 = 8 waves, one 16-wide
// d-tile per wave. K=256 loop of v_wmma_f32_16x16x4_f32.
// ---------------------------------------------------------------------------
__global__ void af_attn_out(const float* __restrict__ text,
                            const float* __restrict__ video,
                            const float* __restrict__ W,
                            float* __restrict__ out) {
  const int b    = blockIdx.z;
  const int mt   = blockIdx.y;
  const int lane = threadIdx.x & 31;
  const int wave = threadIdx.x >> 5;
  const int d0   = (blockIdx.x * 8 + wave) * 16;
  const int q0   = mt * 16;

  const float* Tb = text  + (size_t)b * T_ROWS * D_DIM;
  const float* Vb = video + (size_t)b * V_ROWS * D_DIM;
  const float* Wb = W     + (size_t)b * V_ROWS * T_ROWS;

  const int mrow = lane & 15;
  const int koff = (lane >> 4) * 2;

  const float* aptr = Wb + (size_t)(q0 + mrow) * T_ROWS + koff;   // A = W tile
  const float* bcol = Tb + (size_t)koff * D_DIM + d0 + mrow;      // B = T tile

  v8f c = {};
#pragma unroll 4
  for (int k = 0; k < T_ROWS; k += 4) {
    v2f a = *(const v2f*)(aptr + k);
    v2f bb;
    bb.x = bcol[(size_t)k * D_DIM];
    bb.y = bcol[(size_t)(k + 1) * D_DIM];
    c = __builtin_amdgcn_wmma_f32_16x16x4_f32(false, a, false, bb,
                                              (short)0, c, false, false);
  }

  float* outb = out + (size_t)b * OUT_ROWS * D_DIM;
#pragma unroll
  for (int r = 0; r < 8; ++r) {
    const int m = r + ((lane >> 4) << 3);
    const int n = lane & 15;
    const int q = q0 + m;
    const float val = c[r] + Vb[(size_t)q * D_DIM + d0 + n];
    float* dst = outb + (size_t)(320 + q * REPEATN) * D_DIM + d0 + n;
#pragma unroll
    for (int rep = 0; rep < REPEATN; ++rep) {
      *dst = val;
      dst += D_DIM;
    }
  }
}

// ---------------------------------------------------------------------------
extern "C" void kernel_launch(void* const* d_in, const int* in_sizes, int n_in,
                              void* d_out, int out_size, void* d_ws, size_t ws_size,
                              hipStream_t stream) {
  const float* text  = (const float*)d_in[0];   // [4,256,10240]
  const float* video = (const float*)d_in[1];   // [4,64,10240]
  float* out = (float*)d_out;                   // [4,1600,10240]
  float* S   = (float*)d_ws;                    // [4,64,256] scores -> weights

  // Kernel 0: head copy (4*320*2560 float4 elements)
  {
    const size_t total = (size_t)4 * 320 * (D_DIM / 4);
    const int threads = 256;
    const int blocks = (int)((total + threads - 1) / threads);
    af_copy_head<<<blocks, threads, 0, stream>>>(text, video, out);
  }

  // Kernel 1: scores (WMMA f32), tiles: 16 t-tiles x 4 q-tiles x 4 batches
  af_scores<<<dim3(16, 4, 4), 256, 0, stream>>>(text, video, S);

  // Kernel 2: softmax over 256 keys, one block per (b,q) row
  af_softmax<<<256, 256, 0, stream>>>(S);

  // Kernel 3: attention output + fused replication
  af_attn_out<<<dim3(80, 4, 4), 256, 0, stream>>>(text, video, S, out);
}